// GCN_1_38036230373749
// MI455X (gfx1250) — compile-verified
//
#include <hip/hip_runtime.h>
#include <hip/hip_bf16.h>

typedef float v2f __attribute__((ext_vector_type(2)));
typedef float v8f __attribute__((ext_vector_type(8)));

#define NN 100000
#define NE 800000
#define BN_EPS 1e-5f

__device__ __forceinline__ void atomAddF(float* p, float v) {
    unsafeAtomicAdd(p, v);  // -> global_atomic_add_f32
}

// CDNA5 async global->LDS copy, 16B per lane, tracked by ASYNCcnt.
__device__ __forceinline__ void async_copy_b128(unsigned lds_off, const float* gaddr) {
    asm volatile("global_load_async_to_lds_b128 %0, %1, off"
                 :: "v"(lds_off), "v"(gaddr) : "memory");
}

__device__ __forceinline__ void wait_async0() {
#if __has_builtin(__builtin_amdgcn_s_wait_asynccnt)
    __builtin_amdgcn_s_wait_asynccnt(0);
#else
    asm volatile("s_wait_asynccnt 0" ::: "memory");
#endif
}

// ---------------------------------------------------------------------------
// WMMA fp32 GEMM: C[M,F] (+)= A[M,K] @ B[K,F] (+bias) (relu)
// M = NN, K mult of 32, F mult of 64.
// grid(ceil(M/64), F/64), 128 threads = 4 waves.
// Block tile 64x64; wave w owns rows [bx*64+w*16,+16) x all 64 cols
// (4 v8f accumulators sharing one A fragment -> less LDS traffic per WMMA).
// A/B K-tiles double-buffered in LDS, filled with GLOBAL_LOAD_ASYNC_TO_LDS_B128
// (ASYNCcnt), consumed via DS reads feeding V_WMMA_F32_16X16X4_F32.
// Operand layout per CDNA5 ISA 7.12.2: K = vgpr + 2*(lane>=16) for A and B.
// M-edge guards wrap only VMEM/async ops; WMMAs always run with EXEC all-ones.
// ---------------------------------------------------------------------------
template <bool ACC, bool BIAS, bool RELU>
__global__ __launch_bounds__(128)
void gemm_wmma_f32(const float* __restrict__ A, const float* __restrict__ B,
                   const float* __restrict__ bias, float* __restrict__ C,
                   int K, int F)
{
    __shared__ float As[2][64][36];   // row stride 36: conflict-free, 16B aligned
    __shared__ float Bs[2][32][64];

    const int tid     = threadIdx.x;
    const int lane    = tid & 31;
    const int wave    = tid >> 5;
    const int m       = lane & 15;
    const int halfsel = lane >> 4;            // 0: lanes 0-15, 1: lanes 16-31
    const int row0    = blockIdx.x * 64;
    const int col0    = blockIdx.y * 64;
    const int rowW    = row0 + wave * 16;     // this wave's 16 output rows
    const int arow    = wave * 16 + m;

    v8f acc[4];
    #pragma unroll
    for (int n = 0; n < 4; ++n)
        #pragma unroll
        for (int r = 0; r < 8; ++r)
            acc[n][r] = 0.f;
    if (ACC) {
        #pragma unroll
        for (int n = 0; n < 4; ++n)
            #pragma unroll
            for (int r = 0; r < 8; ++r) {
                int grow = rowW + r + halfsel * 8;
                if (grow < NN)
                    acc[n][r] = C[(size_t)grow * F + col0 + n * 16 + m];
            }
    }

    // async stagers: A tile 64x32 (512 float4), B tile 32x64 (512 float4)
    auto stageA = [&](int buf, int k0) {
        #pragma unroll
        for (int i = 0; i < 4; ++i) {
            int idx = tid + i * 128;
            int r   = idx >> 3;
            int c   = (idx & 7) * 4;
            if (row0 + r < NN)
                async_copy_b128((unsigned)(size_t)&As[buf][r][c],
                                &A[(size_t)(row0 + r) * K + k0 + c]);
        }
    };
    auto stageB = [&](int buf, int k0) {
        #pragma unroll
        for (int i = 0; i < 4; ++i) {
            int idx = tid + i * 128;
            int r   = idx >> 4;
            int c   = (idx & 15) * 4;
            async_copy_b128((unsigned)(size_t)&Bs[buf][r][c],
                            &B[(size_t)(k0 + r) * F + col0 + c]);
        }
    };

    stageA(0, 0);
    stageB(0, 0);
    wait_async0();
    __syncthreads();

    const int nblk = K >> 5;
    for (int blk = 0; blk < nblk; ++blk) {
        const int cur = blk & 1;
        if (blk + 1 < nblk) {               // prefetch next K-tile (async)
            stageA(cur ^ 1, (blk + 1) * 32);
            stageB(cur ^ 1, (blk + 1) * 32);
        }
        #pragma unroll
        for (int kk = 0; kk < 8; ++kk) {
            const int kb = kk * 4 + halfsel * 2;
            v2f a;
            a.x = As[cur][arow][kb];
            a.y = As[cur][arow][kb + 1];
            #pragma unroll
            for (int n = 0; n < 4; ++n) {
                v2f b;
                b.x = Bs[cur][kb][n * 16 + m];
                b.y = Bs[cur][kb + 1][n * 16 + m];
                acc[n] = __builtin_amdgcn_wmma_f32_16x16x4_f32(
                    false, a, false, b, (short)0, acc[n], false, false);
            }
        }
        wait_async0();     // next tile landed in LDS
        __syncthreads();   // all waves done reading cur before it is re-staged
    }

    #pragma unroll
    for (int n = 0; n < 4; ++n)
        #pragma unroll
        for (int r = 0; r < 8; ++r) {
            int grow = rowW + r + halfsel * 8;
            if (grow < NN) {
                float v = acc[n][r];
                if (BIAS) v += bias[col0 + n * 16 + m];
                if (RELU) v = fmaxf(v, 0.f);
                C[(size_t)grow * F + col0 + n * 16 + m] = v;
            }
        }
}

// --------------------------- graph / elementwise ---------------------------

__global__ void deg_init(float* __restrict__ dout, float* __restrict__ din, int n) {
    int i = blockIdx.x * blockDim.x + threadIdx.x;
    if (i < n) { dout[i] = 1.f; din[i] = 1.f; }   // self-loop contribution
}

__global__ void deg_accum(const int* __restrict__ src, const int* __restrict__ dst,
                          float* __restrict__ dout, float* __restrict__ din, int e) {
    int i = blockIdx.x * blockDim.x + threadIdx.x;
    if (i < e) {
        atomAddF(&dout[src[i]], 1.f);
        atomAddF(&din[dst[i]], 1.f);
    }
}

__global__ void norm_finalize(float* __restrict__ d, int n) {
    int i = blockIdx.x * blockDim.x + threadIdx.x;
    if (i < n) d[i] = rsqrtf(fmaxf(d[i], 1.f));
}

// agg[i,f] = h[i,f] * ns[i]   (self-loop message initializes agg: no pre-zero)
__global__ void selfloop_init(const float* __restrict__ h, float* __restrict__ agg,
                              const float* __restrict__ ns, int F, size_t total) {
    size_t stride = (size_t)gridDim.x * blockDim.x;
    for (size_t i = (size_t)blockIdx.x * blockDim.x + threadIdx.x; i < total; i += stride) {
        int r = (int)(i / F);
        agg[i] = h[i] * ns[r];
    }
}

// agg[dst,f] += h[src,f] * ns[src]
__global__ void edge_scatter(const float* __restrict__ h, float* __restrict__ agg,
                             const int* __restrict__ src, const int* __restrict__ dst,
                             const float* __restrict__ ns, int F, size_t total) {
    size_t stride = (size_t)gridDim.x * blockDim.x;
    for (size_t i = (size_t)blockIdx.x * blockDim.x + threadIdx.x; i < total; i += stride) {
        int e = (int)(i / F);
        int f = (int)(i - (size_t)e * F);
        int s = src[e], d = dst[e];
        atomAddF(&agg[(size_t)d * F + f], h[(size_t)s * F + f] * ns[s]);
    }
}

__global__ void row_scale(float* __restrict__ agg, const float* __restrict__ nd,
                          int F, size_t total) {
    size_t stride = (size_t)gridDim.x * blockDim.x;
    for (size_t i = (size_t)blockIdx.x * blockDim.x + threadIdx.x; i < total; i += stride)
        agg[i] *= nd[(int)(i / F)];
}

__global__ void zero_f32(float* __restrict__ p, int n) {
    int i = blockIdx.x * blockDim.x + threadIdx.x;
    if (i < n) p[i] = 0.f;
}

// per-channel sum / sumsq (coalesced rows, register accum, one atomic per block)
__global__ __launch_bounds__(256)
void bn_stats(const float* __restrict__ y, float* __restrict__ sum,
              float* __restrict__ sumsq, int N, int F, int rowsPerBlock) {
    int c0 = threadIdx.x, c1 = threadIdx.x + 256;
    int r0 = blockIdx.x * rowsPerBlock;
    int r1 = min(r0 + rowsPerBlock, N);
    float s0 = 0.f, q0 = 0.f, s1 = 0.f, q1 = 0.f;
    for (int r = r0; r < r1; ++r) {
        const float* row = y + (size_t)r * F;
        if (c0 < F) { float v = row[c0]; s0 += v; q0 += v * v; }
        if (c1 < F) { float v = row[c1]; s1 += v; q1 += v * v; }
    }
    if (c0 < F) { atomAddF(&sum[c0], s0); atomAddF(&sumsq[c0], q0); }
    if (c1 < F) { atomAddF(&sum[c1], s1); atomAddF(&sumsq[c1], q1); }
}

__global__ void bn_finalize(const float* __restrict__ sum, const float* __restrict__ sumsq,
                            float* __restrict__ mean, float* __restrict__ rstd,
                            int F, float invN) {
    int c = blockIdx.x * blockDim.x + threadIdx.x;
    if (c < F) {
        float m = sum[c] * invN;
        float v = sumsq[c] * invN - m * m;
        mean[c] = m;
        rstd[c] = rsqrtf(v + BN_EPS);
    }
}

// y = (y-mean)*rstd*g + beta [+ add] [relu]   (GCN bias b cancels through BN)
__global__ void bn_apply(float* __restrict__ y, const float* __restrict__ mean,
                         const float* __restrict__ rstd, const float* __restrict__ g,
                         const float* __restrict__ beta, const float* __restrict__ add,
                         int F, int relu, size_t total) {
    size_t stride = (size_t)gridDim.x * blockDim.x;
    for (size_t i = (size_t)blockIdx.x * blockDim.x + threadIdx.x; i < total; i += stride) {
        int c = (int)(i % F);
        float v = (y[i] - mean[c]) * rstd[c] * g[c] + beta[c];
        if (add) v += add[i];
        if (relu) v = fmaxf(v, 0.f);
        y[i] = v;
    }
}

// out[node] = dot(h[node,0:128], w) + b   (wave per node, shfl reduction)
__global__ __launch_bounds__(256)
void head_out(const float* __restrict__ h, const float* __restrict__ w,
              const float* __restrict__ b, float* __restrict__ out, int n) {
    int node = blockIdx.x * 8 + (threadIdx.x >> 5);
    int lane = threadIdx.x & 31;
    if (node >= n) return;
    const float* row = h + (size_t)node * 128;
    float s = 0.f;
    #pragma unroll
    for (int j = 0; j < 4; ++j) s += row[lane + j * 32] * w[lane + j * 32];
    #pragma unroll
    for (int off = 16; off > 0; off >>= 1) s += __shfl_xor(s, off, 32);
    if (lane == 0) out[node] = s + b[0];
}

// ------------------------------- host side ---------------------------------

static inline int gblocks(size_t total, int bs = 256, int cap = 262144) {
    size_t b = (total + (size_t)bs - 1) / bs;
    return (int)(b < (size_t)cap ? b : (size_t)cap);
}

extern "C" void kernel_launch(void* const* d_in, const int* in_sizes, int n_in,
                              void* d_out, int out_size, void* d_ws, size_t ws_size,
                              hipStream_t stream) {
    const float* x   = (const float*)d_in[0];
    const float* W1  = (const float*)d_in[1];
    const float* W2  = (const float*)d_in[3];
    const float* W3  = (const float*)d_in[5];
    const float* W4  = (const float*)d_in[7];
    const float* sW1 = (const float*)d_in[9];
    const float* sb1 = (const float*)d_in[10];
    const float* sW2 = (const float*)d_in[11];
    const float* sb2 = (const float*)d_in[12];
    const float* sW3 = (const float*)d_in[13];
    const float* sb3 = (const float*)d_in[14];
    const float* g1  = (const float*)d_in[15];
    const float* be1 = (const float*)d_in[16];
    const float* g2  = (const float*)d_in[17];
    const float* be2 = (const float*)d_in[18];
    const float* g3  = (const float*)d_in[19];
    const float* be3 = (const float*)d_in[20];
    const float* g4  = (const float*)d_in[21];
    const float* be4 = (const float*)d_in[22];
    const float* a1W = (const float*)d_in[23];
    const float* a1b = (const float*)d_in[24];
    const float* a2W = (const float*)d_in[25];
    const float* a2b = (const float*)d_in[26];
    const float* c1W = (const float*)d_in[27];
    const float* c1b = (const float*)d_in[28];
    const float* c2W = (const float*)d_in[29];
    const float* c2b = (const float*)d_in[30];
    const int*   src = (const int*)d_in[31];
    const int*   dst = (const int*)d_in[32];
    float*       out = (float*)d_out;

    float* ws    = (float*)d_ws;
    float* nsrc  = ws;                 // degree -> rsqrt norm (src)
    float* ndst  = ws + NN;            // degree -> rsqrt norm (dst)
    float* ssum  = ws + 2 * NN;        // 512
    float* ssq   = ssum + 512;
    float* smean = ssq + 512;
    float* srstd = smean + 512;
    float* S1    = ws + 2 * NN + 4096; // three 100k x 512 arenas
    float* S2    = S1 + (size_t)NN * 512;
    float* S3    = S2 + (size_t)NN * 512;

    const int EB = 256;
    const int MB = (NN + 63) / 64;                 // 1563 row-blocks of 64
    const int STAT_BLOCKS = 400, STAT_ROWS = 250;  // 400*250 == NN

    #define GEMM(ACCF, BIASF, RELUF, Ap, Bp, BIASP, Cp, Kv, Fv)                          \
        gemm_wmma_f32<ACCF, BIASF, RELUF>                                                \
            <<<dim3(MB, (Fv) / 64), 128, 0, stream>>>((Ap), (Bp), (BIASP), (Cp),         \
                                                      (Kv), (Fv))

    #define AGGREGATE(hbuf, abuf, Fv)                                                    \
        do {                                                                             \
            size_t tn = (size_t)NN * (Fv);                                               \
            size_t te = (size_t)NE * (Fv);                                               \
            selfloop_init<<<gblocks(tn), EB, 0, stream>>>(hbuf, abuf, nsrc, Fv, tn);     \
            edge_scatter<<<gblocks(te), EB, 0, stream>>>(hbuf, abuf, src, dst, nsrc,     \
                                                         Fv, te);                        \
            row_scale<<<gblocks(tn), EB, 0, stream>>>(abuf, ndst, Fv, tn);               \
        } while (0)

    #define BNSTATS(ybuf, Fv)                                                            \
        do {                                                                             \
            zero_f32<<<4, 256, 0, stream>>>(ssum, 1024);                                 \
            bn_stats<<<STAT_BLOCKS, 256, 0, stream>>>(ybuf, ssum, ssq, NN, Fv,           \
                                                      STAT_ROWS);                        \
            bn_finalize<<<2, 256, 0, stream>>>(ssum, ssq, smean, srstd, Fv,              \
                                               1.f / (float)NN);                         \
        } while (0)

    // degree normalization (with self loops)
    deg_init<<<gblocks(NN), EB, 0, stream>>>(nsrc, ndst, NN);
    deg_accum<<<gblocks(NE), EB, 0, stream>>>(src, dst, nsrc, ndst, NE);
    norm_finalize<<<gblocks(NN), EB, 0, stream>>>(nsrc, NN);
    norm_finalize<<<gblocks(NN), EB, 0, stream>>>(ndst, NN);

    // ---- layer 1 (weight first: 128 -> 64) ----
    GEMM(false, false, false, x, W1, (const float*)nullptr, S1, 128, 64);   // t1 = x@W1
    AGGREGATE(S1, S2, 64);                                                  // y1 in S2
    BNSTATS(S2, 64);
    bn_apply<<<gblocks((size_t)NN * 64), EB, 0, stream>>>(
        S2, smean, srstd, g1, be1, (const float*)nullptr, 64, 0, (size_t)NN * 64);
    GEMM(true, true, true, x, sW1, sb1, S2, 128, 64);                       // x1 = relu(bn + x@sW1+sb1)

    // ---- layer 2 (64 -> 256) ----
    AGGREGATE(S2, S1, 64);                                                  // agg2 in S1
    GEMM(false, false, false, S1, W2, (const float*)nullptr, S3, 64, 256);  // t2 in S3
    BNSTATS(S3, 256);
    bn_apply<<<gblocks((size_t)NN * 256), EB, 0, stream>>>(
        S3, smean, srstd, g2, be2, (const float*)nullptr, 256, 0, (size_t)NN * 256);
    GEMM(true, true, true, S2, sW2, sb2, S3, 64, 256);                      // x2 in S3

    // ---- layer 3 (256 -> 512) ----
    AGGREGATE(S3, S1, 256);                                                 // agg3 in S1
    GEMM(false, false, false, S1, W3, (const float*)nullptr, S2, 256, 512); // t3 in S2
    BNSTATS(S2, 512);
    bn_apply<<<gblocks((size_t)NN * 512), EB, 0, stream>>>(
        S2, smean, srstd, g3, be3, (const float*)nullptr, 512, 0, (size_t)NN * 512);
    GEMM(true, true, true, S3, sW3, sb3, S2, 256, 512);                     // x3 in S2

    // ---- layer 4 (512 -> 512, identity skip) ----
    AGGREGATE(S2, S1, 512);                                                 // agg4 in S1
    GEMM(false, false, false, S1, W4, (const float*)nullptr, S3, 512, 512); // t4 in S3
    BNSTATS(S3, 512);
    bn_apply<<<gblocks((size_t)NN * 512), EB, 0, stream>>>(
        S3, smean, srstd, g4, be4, S2 /* x3 */, 512, 1, (size_t)NN * 512);  // x4 in S3

    // ---- heads ----
    GEMM(false, true, true, S3, a1W, a1b, S1, 512, 128);                    // ha
    head_out<<<NN / 8, 256, 0, stream>>>(S1, a2W, a2b, out, NN);
    GEMM(false, true, true, S3, c1W, c1b, S2, 512, 128);                    // hc
    head_out<<<NN / 8, 256, 0, stream>>>(S2, c2W, c2b, out + NN, NN);

    #undef GEMM
    #undef AGGREGATE
    #undef BNSTATS
    (void)in_sizes; (void)n_in; (void)out_size; (void)ws_size;
}